// GNN_73083163509338
// MI455X (gfx1250) — compile-verified
//
#include <hip/hip_runtime.h>
#include <hip/hip_bf16.h>

// ---------------------------------------------------------------------------
// GNN (3-layer GCN, dense-linear via V_WMMA_F32_16X16X4_F32, sparse aggregate)
// B=4, N=4096, C=128, E=131072, L=3
// ---------------------------------------------------------------------------

static constexpr int kB = 4;
static constexpr int kN = 4096;
static constexpr int kC = 128;
static constexpr int kE = 131072;
static constexpr int kMAXDEG = 256;          // mean degree ~64; Poisson tail safe
static constexpr int kBitWordsPerRow = kN / 32;   // 128 u32 per adjacency row

typedef __attribute__((ext_vector_type(2))) float v2f;
typedef __attribute__((ext_vector_type(8))) float v8f;

// ---------------------------------------------------------------------------
// Zero a u32 buffer
// ---------------------------------------------------------------------------
__global__ void k_zero_u32(unsigned int* __restrict__ p, int n) {
    int i = blockIdx.x * blockDim.x + threadIdx.x;
    if (i < n) p[i] = 0u;
}

// ---------------------------------------------------------------------------
// Build symmetric 0/1 adjacency bitmask (diagonal excluded, duplicates dedupe
// automatically via atomicOr -> matches .at[].set(1.0) semantics).
// ---------------------------------------------------------------------------
__global__ void k_build_adj(const int* __restrict__ e0, const int* __restrict__ e1,
                            unsigned int* __restrict__ bits) {
    int e = blockIdx.x * blockDim.x + threadIdx.x;
    if (e >= kE) return;
    int r = e0[e], c = e1[e];
    if (r == c) return;
    atomicOr(&bits[(unsigned)r * kBitWordsPerRow + (c >> 5)], 1u << (c & 31));
    atomicOr(&bits[(unsigned)c * kBitWordsPerRow + (r >> 5)], 1u << (r & 31));
}

// ---------------------------------------------------------------------------
// Compact bitmask rows -> fixed-capacity neighbor lists (deterministic order).
// One thread per row; trivial work (N * 128 words total).
// ---------------------------------------------------------------------------
__global__ void k_compact(const unsigned int* __restrict__ bits,
                          int* __restrict__ cnt, int* __restrict__ nbr) {
    int n = blockIdx.x * blockDim.x + threadIdx.x;
    if (n >= kN) return;
    const unsigned int* row = bits + (size_t)n * kBitWordsPerRow;
    int* out = nbr + (size_t)n * kMAXDEG;
    int c = 0;
    for (int w = 0; w < kBitWordsPerRow; ++w) {
        unsigned int v = row[w];
        while (v) {
            int t = __builtin_ctz(v);
            v &= v - 1;
            if (c < kMAXDEG) out[c++] = (w << 5) + t;
        }
    }
    cnt[n] = c;
}

// ---------------------------------------------------------------------------
// Sparse neighbor aggregation: agg[b,n,:] = sum_{m in nbr(n)} h[b,m,:]
// One 128-thread block per (b,n); neighbor list staged through LDS.
// ---------------------------------------------------------------------------
__global__ __launch_bounds__(kC) void k_aggregate(const float* __restrict__ h,
                                                  const int* __restrict__ cnt,
                                                  const int* __restrict__ nbr,
                                                  float* __restrict__ agg) {
    __shared__ int lst[kMAXDEG];
    int bn = blockIdx.x;                // b*N + n
    int n  = bn & (kN - 1);
    int b  = bn >> 12;
    int t  = threadIdx.x;               // channel
    int deg = cnt[n];
    const int* g = nbr + (size_t)n * kMAXDEG;
    for (int j = t; j < deg; j += kC) lst[j] = g[j];
    __syncthreads();
    float acc = 0.0f;
    const float* hb = h + (size_t)b * kN * kC + t;
    for (int j = 0; j < deg; ++j)
        acc += hb[(size_t)lst[j] * kC];
    agg[(size_t)bn * kC + t] = acc;
}

// ---------------------------------------------------------------------------
// WMMA fp32 GEMM: out[M x 128] = epi( concat(A0,A1)[M x K] * W[128 x K]^T + b )
//   K = 128 (A1 unused) or 256 (A0 = first 128 cols, A1 = last 128 cols)
//   epi: optional exact-erf GELU, optional residual add.
// One wave per 16x16 tile; block = 8 waves = one row-tile across all 8 col-tiles.
// M = B*N = 16384 (divisible by 16) -> no bounds checks, EXEC all-ones (WMMA req).
//
// V_WMMA_F32_16X16X4_F32 fragment layouts (ISA 7.12.2):
//   A (16x4): lane l -> row m0+(l&15), ks k0+2*(l>>4) + {0,1}   => float2 load
//   B (4x16): lane l -> col n0+(l&15), ks k0+2*(l>>4) + {0,1}   => float2 from W
//   C/D     : acc[v]  -> row m0 + 8*(l>>4) + v, col n0 + (l&15)
// ---------------------------------------------------------------------------
template <int K, bool GELU, bool RES>
__global__ __launch_bounds__(256) void gemm_wmma_f32(
    const float* __restrict__ A0, const float* __restrict__ A1,
    const float* __restrict__ W,  const float* __restrict__ bias,
    const float* __restrict__ res, float* __restrict__ out) {
    const int lane = threadIdx.x & 31;
    const int lr = lane & 15;
    const int hi = lane >> 4;
    const int m0 = blockIdx.x << 4;
    const int n0 = ((threadIdx.x >> 5) & 7) << 4;

    const float* arow = A0 + (size_t)(m0 + lr) * kC + 2 * hi;
    const float* wrow = W  + (size_t)(n0 + lr) * K  + 2 * hi;

    v8f acc = {};
#pragma unroll
    for (int k0 = 0; k0 < 128; k0 += 4) {
        v2f a = *(const v2f*)(arow + k0);
        v2f b = *(const v2f*)(wrow + k0);
        acc = __builtin_amdgcn_wmma_f32_16x16x4_f32(
            false, a, false, b, (short)0, acc, false, false);
    }
    if (K == 256) {
        const float* arow1 = A1 + (size_t)(m0 + lr) * kC + 2 * hi;
#pragma unroll
        for (int k0 = 0; k0 < 128; k0 += 4) {
            v2f a = *(const v2f*)(arow1 + k0);
            v2f b = *(const v2f*)(wrow + 128 + k0);
            acc = __builtin_amdgcn_wmma_f32_16x16x4_f32(
                false, a, false, b, (short)0, acc, false, false);
        }
    }

    const int col = n0 + lr;
    const float bv = bias[col];
#pragma unroll
    for (int v = 0; v < 8; ++v) {
        int row = m0 + hi * 8 + v;
        float val = acc[v] + bv;
        if (GELU) val = 0.5f * val * (1.0f + erff(val * 0.70710678118654752f));
        if (RES) val += res[(size_t)row * kC + col];
        out[(size_t)row * kC + col] = val;
    }
}

// ---------------------------------------------------------------------------
// v_loc: out[bn] = dot(hh[bn,:], W2[0,:]) + b2[0]   (128-wide dot, LDS reduce)
// ---------------------------------------------------------------------------
__global__ __launch_bounds__(kC) void k_vloc(const float* __restrict__ hh,
                                             const float* __restrict__ W2,
                                             const float* __restrict__ b2,
                                             float* __restrict__ out) {
    __shared__ float red[kC];
    int row = blockIdx.x;
    int t = threadIdx.x;
    red[t] = hh[(size_t)row * kC + t] * W2[t];
    __syncthreads();
    for (int s = 64; s > 0; s >>= 1) {
        if (t < s) red[t] += red[t + s];
        __syncthreads();
    }
    if (t == 0) out[row] = red[0] + b2[0];
}

// ---------------------------------------------------------------------------
// means[:,1:]: out[(b*(N-1)+(n-1))*3 + d] = dot(x[b,n,:], out_W[d,:]) + out_b[d]
// ---------------------------------------------------------------------------
__global__ __launch_bounds__(kC) void k_means(const float* __restrict__ x,
                                              const float* __restrict__ Wo,
                                              const float* __restrict__ bo,
                                              float* __restrict__ out) {
    int bn = blockIdx.x;
    int n = bn & (kN - 1);
    int b = bn >> 12;
    if (n == 0) return;  // uniform per-block: slice means[:, 1:]
    __shared__ float red[3][kC];
    int t = threadIdx.x;
    float xv = x[(size_t)bn * kC + t];
    red[0][t] = xv * Wo[0 * kC + t];
    red[1][t] = xv * Wo[1 * kC + t];
    red[2][t] = xv * Wo[2 * kC + t];
    __syncthreads();
    for (int s = 64; s > 0; s >>= 1) {
        if (t < s) {
            red[0][t] += red[0][t + s];
            red[1][t] += red[1][t + s];
            red[2][t] += red[2][t + s];
        }
        __syncthreads();
    }
    if (t < 3)
        out[((size_t)b * (kN - 1) + (n - 1)) * 3 + t] = red[t][0] + bo[t];
}

// ---------------------------------------------------------------------------
// Host-side orchestration
// ---------------------------------------------------------------------------
extern "C" void kernel_launch(void* const* d_in, const int* in_sizes, int n_in,
                              void* d_out, int out_size, void* d_ws, size_t ws_size,
                              hipStream_t stream) {
    const float* x          = (const float*)d_in[0];
    const float* lin0_W     = (const float*)d_in[1];
    const float* lin0_b     = (const float*)d_in[2];
    const float* lin1_W     = (const float*)d_in[3];
    const float* lin1_b     = (const float*)d_in[4];
    const float* lin2_W     = (const float*)d_in[5];
    const float* lin2_b     = (const float*)d_in[6];
    const float* outc_W     = (const float*)d_in[7];
    const float* outc_b     = (const float*)d_in[8];
    const float* head_W1    = (const float*)d_in[9];
    const float* head_b1    = (const float*)d_in[10];
    const float* head_W2    = (const float*)d_in[11];
    const float* head_b2    = (const float*)d_in[12];
    const float* out_W      = (const float*)d_in[13];
    const float* out_b      = (const float*)d_in[14];
    const int*   edge_index = (const int*)d_in[15];

    const int* e0 = edge_index;
    const int* e1 = edge_index + kE;

    // ---- workspace carve-up ------------------------------------------------
    char* ws = (char*)d_ws;
    unsigned int* adj_bits = (unsigned int*)ws;                    // 2 MB
    size_t off = (size_t)kN * kBitWordsPerRow * sizeof(unsigned int);
    int* nbr_cnt = (int*)(ws + off);                               // 16 KB
    off += (size_t)kN * sizeof(int);
    int* nbr_list = (int*)(ws + off);                              // 4 MB
    off += (size_t)kN * kMAXDEG * sizeof(int);
    const size_t feat = (size_t)kB * kN * kC;                      // 2M floats
    float* xbuf = (float*)(ws + off);  off += feat * sizeof(float);
    float* hbuf = (float*)(ws + off);  off += feat * sizeof(float);
    float* aggb = (float*)(ws + off);  off += feat * sizeof(float);
    float* t1b  = (float*)(ws + off);  off += feat * sizeof(float);

    float* vloc_out  = (float*)d_out;                  // [B,N,1] = 16384
    float* means_out = (float*)d_out + (size_t)kB * kN;

    const int nBitWords = kN * kBitWordsPerRow;
    const int gemmGrid = (kB * kN) / 16;               // 1024 row tiles
    const int bnGrid = kB * kN;                        // 16384

    // ---- adjacency (rebuilt every call; idempotent & deterministic) --------
    k_zero_u32<<<(nBitWords + 255) / 256, 256, 0, stream>>>(adj_bits, nBitWords);
    k_build_adj<<<(kE + 255) / 256, 256, 0, stream>>>(e0, e1, adj_bits);
    k_compact<<<(kN + 127) / 128, 128, 0, stream>>>(adj_bits, nbr_cnt, nbr_list);

    // ---- x working copy ----------------------------------------------------
    hipMemcpyAsync(xbuf, x, feat * sizeof(float), hipMemcpyDeviceToDevice, stream);

    auto gcn_layer = [&](int l) {
        const float* W0 = lin0_W + (size_t)l * kC * kC;
        const float* b0 = lin0_b + (size_t)l * kC;
        const float* W1 = lin1_W + (size_t)l * kC * (2 * kC);
        const float* b1 = lin1_b + (size_t)l * kC;
        const float* W2 = lin2_W + (size_t)l * kC * kC;
        const float* b2 = lin2_b + (size_t)l * kC;
        const float* Wo = outc_W + (size_t)l * kC * kC;
        const float* bo = outc_b + (size_t)l * kC;

        // h = x @ W0^T + b0
        gemm_wmma_f32<128, false, false><<<gemmGrid, 256, 0, stream>>>(
            xbuf, nullptr, W0, b0, nullptr, hbuf);
        // agg = A * h (sparse)
        k_aggregate<<<bnGrid, kC, 0, stream>>>(hbuf, nbr_cnt, nbr_list, aggb);
        // t1 = [x, agg] @ W1^T + b1
        gemm_wmma_f32<256, false, false><<<gemmGrid, 256, 0, stream>>>(
            xbuf, aggb, W1, b1, nullptr, t1b);
        // t2 = gelu(t1 @ W2^T + b2)   (reuse hbuf)
        gemm_wmma_f32<128, true, false><<<gemmGrid, 256, 0, stream>>>(
            t1b, nullptr, W2, b2, nullptr, hbuf);
        // x = x + t2 @ Wo^T + bo      (in-place residual: 1 read + 1 write per elem)
        gemm_wmma_f32<128, false, true><<<gemmGrid, 256, 0, stream>>>(
            hbuf, nullptr, Wo, bo, xbuf, xbuf);
    };

    gcn_layer(0);
    gcn_layer(1);

    // ---- head (on x after 2 layers) ---------------------------------------
    gemm_wmma_f32<128, true, false><<<gemmGrid, 256, 0, stream>>>(
        xbuf, nullptr, head_W1, head_b1, nullptr, t1b);
    k_vloc<<<bnGrid, kC, 0, stream>>>(t1b, head_W2, head_b2, vloc_out);

    // ---- final layer + means ----------------------------------------------
    gcn_layer(2);
    k_means<<<bnGrid, kC, 0, stream>>>(xbuf, out_W, out_b, means_out);

    (void)in_sizes; (void)n_in; (void)out_size; (void)ws_size; (void)off;
}